// InverseCovMatrixFromPositions_4449586119492
// MI455X (gfx1250) — compile-verified
//
#include <hip/hip_runtime.h>
#include <hip/hip_bf16.h>

// Batched 20x20 SPD inverse: cov = sigma^2*exp(-phi*dist) + tau^2*I, out = inv(cov).
// One matrix per 20 threads (column-owner layout), 16 matrices per 320-thread block.
// In-place Gauss-Jordan without pivoting (matrix is SPD with nugget).

namespace {
constexpr int   kN       = 20;            // matrix dim / neighbor size
constexpr int   kMPB     = 16;            // matrices per block
constexpr int   kTPB     = kMPB * kN;     // 320 threads = 10 wave32, no idle lanes
constexpr int   kCStride = 24;            // LDS column stride (floats); 96B -> 16B aligned
constexpr int   kMStride = kCStride * kN; // 480 floats per matrix (bank offset 32 between matrices)
constexpr float kPhi     = 1.5f;
constexpr float kSigmaSq = 1.0f;
constexpr float kTauSq   = 0.01f * 1.0f;  // TAU * SIGMA_SQ
}

__global__ __launch_bounds__(kTPB) void inv_cov_gj_kernel(
    const float* __restrict__ pos, float* __restrict__ out) {
  __shared__ __align__(16) float A[kMPB * kMStride];   // 30 KB: matrices, column-major
  __shared__ __align__(16) float P[kMPB * kN * 2];     // 2.5 KB: positions

  const int t = threadIdx.x;
  const int m = t / kN;          // matrix within block
  const int j = t - m * kN;      // owned column

  // ---- Stage this block's 16*20 float2 positions into LDS (2560 B, coalesced) ----
  {
    const float* g = pos + ((long long)blockIdx.x * (kMPB * kN) + t) * 2;
#if defined(__gfx1250__)
    // CDNA5 async global->LDS copy (ASYNCcnt path). VDST = LDS byte address
    // (low 32 bits of the generic pointer, per the ISA aperture rule),
    // VADDR = 64-bit global address, "off" = no SADDR (GV mode).
    const unsigned           lds_addr = (unsigned)(size_t)&P[t * 2];
    const unsigned long long gaddr    = (unsigned long long)(size_t)g;
    asm volatile("global_load_async_to_lds_b64 %0, %1, off"
                 :
                 : "v"(lds_addr), "v"(gaddr)
                 : "memory");
    asm volatile("s_wait_asynccnt 0" ::: "memory");
#else
    const float2 v = *(const float2*)g;
    *(float2*)&P[t * 2] = v;
#endif
  }
  __syncthreads();

  // ---- Build covariance column j in registers ----
  const float2* Pm = (const float2*)&P[m * kN * 2];
  const float2 pj = Pm[j];
  float c[kN];
#pragma unroll
  for (int i = 0; i < kN; ++i) {
    const float2 pi = Pm[i];                 // broadcast read within each matrix
    const float dx = pi.x - pj.x;
    const float dy = pi.y - pj.y;
    const float d  = sqrtf(dx * dx + dy * dy);
    const float cv = kSigmaSq * __expf(-kPhi * d);
    c[i] = (i == j) ? (cv + kTauSq) : cv;
  }

  float* __restrict__ Acol = &A[m * kMStride + j * kCStride];
  const float* __restrict__ Amat = &A[m * kMStride];

  // publish initial columns so step k=0 can read column 0
#pragma unroll
  for (int i = 0; i < kN; ++i) Acol[i] = c[i];
  __syncthreads();

  // ---- In-place Gauss-Jordan, fully unrolled (static register indices) ----
  // Per step k:
  //   colk[i] = A[i][k]  (broadcast 5x ds_load_b128)
  //   inv = 1/A[k][k]; s_j = (j==k) ? inv : A[k][j]*inv
  //   i!=k: A[i][j] <- (j==k ? 0 : A[i][j]) - colk[i]*s_j ;  A[k][j] <- s_j
  // Only the owner of next step's pivot column (j==k+1) writes LDS -> one barrier/step.
#pragma unroll
  for (int k = 0; k < kN; ++k) {
    float colk[kN];
#pragma unroll
    for (int i = 0; i < kN; ++i) colk[i] = Amat[k * kCStride + i];

    const float inv  = 1.0f / colk[k];
    const bool  diag = (j == k);
    const float s    = diag ? inv : c[k] * inv;
#pragma unroll
    for (int i = 0; i < kN; ++i) {
      if (i != k) {
        const float base = diag ? 0.0f : c[i];
        c[i] = fmaf(-colk[i], s, base);
      }
    }
    c[k] = s;

    if (k + 1 < kN) {
      if (j == k + 1) {
#pragma unroll
        for (int i = 0; i < kN; ++i) Acol[i] = c[i];
      }
    }
    __syncthreads();
  }

  // ---- Publish final columns, then coalesced float4 output ----
  // inv(cov) is symmetric, so the column-major LDS image equals the row-major result.
#pragma unroll
  for (int i = 0; i < kN; ++i) Acol[i] = c[i];
  __syncthreads();

  float4* __restrict__ gout =
      (float4*)(out + (long long)blockIdx.x * (kMPB * kN * kN));
#pragma unroll
  for (int r = 0; r < (kMPB * kN * kN) / (4 * kTPB); ++r) {  // 5 iterations
    const int f4  = r * kTPB + t;            // float4 index within block chunk
    const int q   = f4 * 4;                  // flat float index (0..6396, %4==0)
    const int m2  = q / (kN * kN);
    const int off = q - m2 * (kN * kN);      // 0..396
    const int col = off / kN;                // LDS column
    const int i0  = off - col * kN;          // 0,4,8,12,16 -> never crosses a column
    const float4 v = *(const float4*)&A[m2 * kMStride + col * kCStride + i0];
    gout[f4] = v;
  }
}

extern "C" void kernel_launch(void* const* d_in, const int* in_sizes, int n_in,
                              void* d_out, int out_size, void* d_ws, size_t ws_size,
                              hipStream_t stream) {
  const float* pos = (const float*)d_in[0];
  // d_in[1] (edge_list) is unused by the reference output.
  float* out = (float*)d_out;

  const long long B    = (long long)in_sizes[0] / (kN * 2);  // 131072 matrices
  const int       grid = (int)(B / kMPB);                    // 8192 full blocks (B % 16 == 0)
  inv_cov_gj_kernel<<<grid, kTPB, 0, stream>>>(pos, out);
}